// InvButterflyLayer_7868380086486
// MI455X (gfx1250) — compile-verified
//
#include <hip/hip_runtime.h>
#include <hip/hip_bf16.h>

typedef __attribute__((ext_vector_type(2))) float v2f;
typedef __attribute__((ext_vector_type(8))) float v8f;

#define B_     256
#define MID_   16384
#define C_     16
#define NLVL_  8
#define NB_    256
#define ROWS_  512
#define RSTR_  17                 // padded LDS row stride (floats), gcd(17,64)=1
#define CHAN_  (ROWS_ * RSTR_)    // floats per channel (8704)
#define HALFB_ (2 * CHAN_)        // floats per ping-pong buffer (17408)

__device__ __forceinline__ v8f wmma4(v2f a, v2f b, v8f c) {
  // D = A(16x4, f32) * B(4x16, f32) + C(16x16, f32)
  return __builtin_amdgcn_wmma_f32_16x16x4_f32(false, a, false, b, (short)0, c, false, false);
}

__global__ __launch_bounds__(256)
void invbfly_kernel(const float* __restrict__ in_data,    // (B, MID, 2)
                    const float* __restrict__ mid_dense,  // (1, MID, 2)
                    const float* __restrict__ in_filter,  // (64, 1, 16)
                    const float* __restrict__ in_bias,    // (16,)
                    const float* __restrict__ filters,    // (8, 256, 2, 16, 16)
                    const float* __restrict__ biases,     // (8, 256, 16)
                    const float* __restrict__ fea_dense,  // (256, 16, 64)
                    float* __restrict__ out) {            // [B*8192 out_data][B*8192 out1]
  __shared__ float lds[2 * HALFB_];                       // two ping-pong buffers
  const int b    = blockIdx.x;
  const int tid  = threadIdx.x;
  const int lane = tid & 31;
  const int wave = tid >> 5;
  const int col  = lane & 15;   // M-row index for A frag, N-col for B/C/D frags
  const int hi   = lane >> 4;   // 0: lanes 0-15 (K+0/K+1), 1: lanes 16-31 (K+2/K+3)

  // ---------------- Stage A: input conv, s[i,pos,c], pos in [0,511) ----------------
  {
    const float bias = in_bias[col];
    for (int task = wave; task < 64; task += 8) {      // 2 channels * 32 M-tiles
      const int i  = task >> 5;
      const int t0 = (task & 31) << 4;
      v8f acc;
      #pragma unroll
      for (int r = 0; r < 8; ++r) acc[r] = bias;
      const int mbase = (t0 + col) * 32;               // window start for this A-row
      #pragma unroll
      for (int k = 0; k < 16; ++k) {                   // K = 64 -> 16 steps of 4
        const int r0 = 4 * k + (hi << 1);
        int g0 = mbase + r0;     if (g0 > MID_ - 1) g0 = MID_ - 1;   // clamp (tile 31 row 15 is invalid anyway)
        int g1 = mbase + r0 + 1; if (g1 > MID_ - 1) g1 = MID_ - 1;
        v2f a, w;
        a.x = in_data[(b * MID_ + g0) * 2 + i] * mid_dense[g0 * 2 + i];
        a.y = in_data[(b * MID_ + g1) * 2 + i] * mid_dense[g1 * 2 + i];
        w.x = in_filter[r0 * 16 + col];
        w.y = in_filter[(r0 + 1) * 16 + col];
        acc = wmma4(a, w, acc);
      }
      float* dst = lds + i * CHAN_;                    // buffer 0
      #pragma unroll
      for (int r = 0; r < 8; ++r) {
        const int t = t0 + r + (hi << 3);
        if (t < 511) dst[t * RSTR_ + col] = fmaxf(acc[r], 0.0f);
      }
    }
  }
  __syncthreads();

  // ---------------- Tree levels 1..8 ----------------
  int Tprev = 511;
  for (int lvl = 1; lvl <= NLVL_; ++lvl) {
    const int nb      = 1 << lvl;
    const int T       = (Tprev - 1) >> 1;              // 255,127,63,31,15,7,3,1
    const int ntshift = (lvl < 5) ? (5 - lvl) : 0;     // log2(ceil(T/16)) — always a pow2 here
    const int ntiles  = 1 << ntshift;
    const float* src  = lds + ((lvl & 1) ? 0 : HALFB_);
    float*       dst  = lds + ((lvl & 1) ? HALFB_ : 0);
    const int ntasks  = 2 * nb * ntiles;
    for (int task = wave; task < ntasks; task += 8) {
      const int tile = task & (ntiles - 1);
      const int rest = task >> ntshift;
      const int n    = rest & (nb - 1);
      const int i    = rest >> lvl;
      const int t0   = tile << 4;
      const float* Wp = filters + ((lvl - 1) * NB_ + n) * 512;     // (2,16,16) row-major = K=32 x N=16
      const float bias = biases[((lvl - 1) * NB_ + n) * C_ + col];
      v8f acc;
      #pragma unroll
      for (int r = 0; r < 8; ++r) acc[r] = bias;
      // A row m = t0+col reads prev rows 2*(t0+col)+k, K index j = k*16 + c
      const float* sA = src + i * CHAN_ + ((n >> 1) * Tprev + 2 * (t0 + col)) * RSTR_;
      #pragma unroll
      for (int k = 0; k < 8; ++k) {                    // K = 32 -> 8 steps of 4
        const int j0 = 4 * k + (hi << 1);
        const int j1 = j0 + 1;
        v2f a, w;
        a.x = sA[(j0 >> 4) * RSTR_ + (j0 & 15)];
        a.y = sA[(j1 >> 4) * RSTR_ + (j1 & 15)];
        w.x = Wp[j0 * 16 + col];
        w.y = Wp[j1 * 16 + col];
        acc = wmma4(a, w, acc);
      }
      float* dA = dst + i * CHAN_ + n * T * RSTR_;
      #pragma unroll
      for (int r = 0; r < 8; ++r) {                    // mask invalid rows at store only
        const int t = t0 + r + (hi << 3);
        if (t < T) dA[t * RSTR_ + col] = fmaxf(acc[r], 0.0f);
      }
    }
    __syncthreads();
    Tprev = T;
  }

  // ---------------- Final projection + R/I recombine ----------------
  // feat[i][n][c] sits in buffer 0 (level 8 output, T=1 -> row n).
  const float* feat = lds;
  const float inv = 1.0f / (float)MID_;
  for (int idx = tid; idx < NB_ * 32; idx += 256) {    // tasks = (n, fe), fe = f/2
    const int fe = idx & 31;
    const int n  = idx >> 5;
    const float* f0 = feat + 0 * CHAN_ + n * RSTR_;
    const float* f1 = feat + 1 * CHAN_ + n * RSTR_;
    const float* fd = fea_dense + n * C_ * 64;
    float d0e = 0.0f, d0o = 0.0f, d1o = 0.0f;
    #pragma unroll
    for (int c = 0; c < 16; ++c) {
      const float we = fd[c * 64 + 2 * fe];
      const float wo = fd[c * 64 + 2 * fe + 1];
      d0e += f0[c] * we;   // outp[b,0, n*64 + 2fe]
      d0o += f0[c] * wo;   // outp[b,0, n*64 + 2fe+1]  -> R[:,:,1]
      d1o += f1[c] * wo;   // outp[b,1, n*64 + 2fe+1]  -> I[:,:,1]
    }
    const int q = n * 32 + fe;                         // [0, 8192)
    out[b * 8192 + q]               = (d0e - d1o) * inv;   // out_data
    out[B_ * 8192 + b * 8192 + q]   = d0o;                 // R[:,:,1]
  }
}

extern "C" void kernel_launch(void* const* d_in, const int* in_sizes, int n_in,
                              void* d_out, int out_size, void* d_ws, size_t ws_size,
                              hipStream_t stream) {
  const float* in_data   = (const float*)d_in[0];
  const float* mid_dense = (const float*)d_in[1];
  const float* in_filter = (const float*)d_in[2];
  const float* in_bias   = (const float*)d_in[3];
  const float* filters   = (const float*)d_in[4];
  const float* biases    = (const float*)d_in[5];
  const float* fea_dense = (const float*)d_in[6];
  (void)in_sizes; (void)n_in; (void)out_size; (void)d_ws; (void)ws_size;
  invbfly_kernel<<<B_, 256, 0, stream>>>(in_data, mid_dense, in_filter, in_bias,
                                         filters, biases, fea_dense, (float*)d_out);
}